// Tree_83193516523595
// MI455X (gfx1250) — compile-verified
//
#include <hip/hip_runtime.h>
#include <hip/hip_bf16.h>

typedef __attribute__((ext_vector_type(2))) float v2f;
typedef __attribute__((ext_vector_type(4))) float v4f;
typedef __attribute__((ext_vector_type(8))) float v8f;

#define TREE_DEPTH 10
#define N_NODES    1024     // 2^10
#define OUT_W      2048     // 2^(DEPTH+1)
#define TB         16       // batch rows per block (one WMMA M tile)
#define NTHREADS   256      // 8 wave32

__device__ __forceinline__ float sigm(float t) {
    // v_exp_f32 + v_add + v_rcp_f32 (avoid the IEEE div_scale/div_fmas sequence)
    return __builtin_amdgcn_rcpf(1.0f + __expf(-t));
}

__global__ __launch_bounds__(NTHREADS)
void tree_forward_kernel(const float* __restrict__ x,
                         const float* __restrict__ W,
                         const float* __restrict__ bias,
                         float* __restrict__ out)
{
    extern __shared__ float smem[];
    float* sX  = smem;                  // 16
    float* sW  = smem + 16;             // 1024
    float* sB  = sW + N_NODES;          // 1024
    float* sD  = sB + N_NODES;          // TB*1024  (sigmoid outputs)
    float* sMu = sD + TB * N_NODES;     // TB*2048  (mu staging)

    const int t    = threadIdx.x;
    const int lane = t & 31;
    const int wave = t >> 5;
    const int b0   = blockIdx.x * TB;

    // ---- Phase 1: stage x tile, W, bias in LDS ----
    if (t < TB) sX[t] = x[b0 + t];
    for (int i = t; i < N_NODES; i += NTHREADS) {
        sW[i] = W[i];
        sB[i] = bias[i];
    }
    __syncthreads();

    // ---- Phase 2: pre-activations via V_WMMA_F32_16X16X4_F32 ----
    // A (16x4): only K=0 column nonzero = x[m];  lanes0-15 VGPR0 hold (M=lane,K=0)
    // B (4x16): only K=0 row nonzero = W[n];     lanes0-15 VGPR0 hold (K=0,N=lane)
    // C (16x16): bias[n] broadcast over M (N = lane&15 for every C VGPR)
    // D[m][n] = x[m]*W[n] + bias[n]  -> sigmoid -> sD[m][n]
    {
        v2f A;
        A.x = (lane < 16) ? sX[lane] : 0.0f;
        A.y = 0.0f;
        #pragma unroll
        for (int j = 0; j < 8; ++j) {
            const int nt = wave * 8 + j;        // node tile 0..63
            const int nb = nt * 16;
            v2f Bm;
            Bm.x = (lane < 16) ? sW[nb + lane] : 0.0f;
            Bm.y = 0.0f;
            const float bv = sB[nb + (lane & 15)];
            v8f C;
            #pragma unroll
            for (int v = 0; v < 8; ++v) C[v] = bv;

            v8f D = __builtin_amdgcn_wmma_f32_16x16x4_f32(
                false, A, false, Bm, (short)0, C, false, false);

            const int n  = nb + (lane & 15);
            const int mb = (lane < 16) ? 0 : 8;  // C/D layout: VGPR v -> M=v (lo lanes), M=v+8 (hi lanes)
            #pragma unroll
            for (int v = 0; v < 8; ++v) {
                sD[(mb + v) * N_NODES + n] = sigm(D[v]);
            }
        }
    }
    __syncthreads();

    // ---- Phase 3: tree expansion into sMu ----
    {
        const int brow = t >> 4;                 // 0..15 batch row
        const int rr   = t & 15;
        const float* __restrict__ sDrow = sD + brow * N_NODES;
        float* __restrict__ sMuRow = sMu + brow * OUT_W;

        if (rr == 0) {
            // shallow nodes 0..31
            sMuRow[0] = 1.0f;
            sMuRow[1] = 1.0f;
            float cur[16];
            cur[0] = 1.0f;
            int base = 1;
            #pragma unroll
            for (int k = 0; k < 4; ++k) {
                const int sz = 1 << k;
                const int cb = base << 1;
                #pragma unroll
                for (int i = sz - 1; i >= 0; --i) {
                    const float d  = sDrow[base + i];
                    const float p  = cur[i];
                    const float c0 = p * d;
                    const float c1 = p * (1.0f - d);
                    cur[2 * i]     = c0;
                    cur[2 * i + 1] = c1;
                    sMuRow[cb + 2 * i]     = c0;
                    sMuRow[cb + 2 * i + 1] = c1;
                }
                base = cb;
            }
        }

        // two depth-5 subtrees per thread: roots 32+rr and 48+rr
        #pragma unroll
        for (int s = 0; s < 2; ++s) {
            const int rr5 = rr + s * 16;
            // path product down to node 32+rr5
            int   n = 1;
            float P = 1.0f;
            #pragma unroll
            for (int dep = 4; dep >= 0; --dep) {
                const int   bit = (rr5 >> dep) & 1;
                const float d   = sDrow[n];
                P *= bit ? (1.0f - d) : d;
                n = (n << 1) + bit;
            }
            // expand subtree (relative depths 0..5 => 63 nodes), all in registers
            float cur[32];
            cur[0] = P;
            sMuRow[n] = P;
            int base = n;
            #pragma unroll
            for (int k = 0; k < 5; ++k) {
                const int sz = 1 << k;
                const int cb = base << 1;
                #pragma unroll
                for (int i = sz - 1; i >= 0; --i) {
                    const float d  = sDrow[base + i];
                    const float p  = cur[i];
                    const float c0 = p * d;
                    const float c1 = p * (1.0f - d);
                    cur[2 * i]     = c0;
                    cur[2 * i + 1] = c1;
                    sMuRow[cb + 2 * i]     = c0;
                    sMuRow[cb + 2 * i + 1] = c1;
                }
                base = cb;
            }
        }
    }
    __syncthreads();

    // ---- Phase 4: coalesced streaming store (non-temporal, never re-read) ----
    {
        const v4f* __restrict__ src = (const v4f*)sMu;
        v4f* __restrict__ dst = (v4f*)(out + (size_t)b0 * OUT_W);
        #pragma unroll 4
        for (int i = t; i < (TB * OUT_W) / 4; i += NTHREADS) {
            __builtin_nontemporal_store(src[i], dst + i);
        }
    }
}

extern "C" void kernel_launch(void* const* d_in, const int* in_sizes, int n_in,
                              void* d_out, int out_size, void* d_ws, size_t ws_size,
                              hipStream_t stream) {
    const float* x    = (const float*)d_in[0];   // [B, 1]
    const float* W    = (const float*)d_in[1];   // [1024, 1, 1]
    const float* bias = (const float*)d_in[2];   // [1024, 1]
    float* out = (float*)d_out;                  // [B, 2048]

    const int B = in_sizes[0];                   // F == 1
    const int blocks = B / TB;                   // 65536/16 = 4096

    const size_t lds_bytes =
        (size_t)(16 + 2 * N_NODES + TB * N_NODES + TB * OUT_W) * sizeof(float); // ~200 KB

    // allow >64KB dynamic LDS (gfx1250: 320 KB per WGP)
    (void)hipFuncSetAttribute((const void*)tree_forward_kernel,
                              hipFuncAttributeMaxDynamicSharedMemorySize,
                              (int)lds_bytes);

    tree_forward_kernel<<<blocks, NTHREADS, lds_bytes, stream>>>(x, W, bias, out);
}